// Recall_53077205844588
// MI455X (gfx1250) — compile-verified
//
#include <hip/hip_runtime.h>

typedef __attribute__((ext_vector_type(4))) float v4f;
typedef __attribute__((ext_vector_type(4))) int   v4i;
typedef __attribute__((ext_vector_type(2))) float v2f;
typedef __attribute__((ext_vector_type(8))) float v8f;

#define B_  32
#define F_  512
#define T_  2048
#define TV_ (T_ / 4)            // 512 float4 vectors per (b,f) row
#define NV_ (B_ * F_ * TV_)     // 8,388,608 total vectors

// Sum a value across all 32 lanes of a wave using V_WMMA_F32_16X16X4_F32.
// A (16x4 f32, 2 VGPRs) holds each lane's partial in VGPR0 and 0 in VGPR1;
// B = all-ones (uniform, so its exact lane layout is irrelevant).
// D[m,n] = sum_k A[m,k]. Per the documented 16x16 f32 C/D layout, summing the
// 8 D VGPRs in lane n yields sum over rows 0..7 (lanes 0-15) or rows 8..15
// (lanes 16-31) of column n; one shfl_xor(16) completes the full 64-slot sum
// (= sum of all 32 lane partials), broadcast to every lane.
// Requires EXEC == all ones (call sites keep full waves active).
__device__ __forceinline__ float wave_sum_wmma(float x) {
  v2f a; a[0] = x;    a[1] = 0.0f;
  v2f b; b[0] = 1.0f; b[1] = 1.0f;
  v8f c = {};
  v8f d = __builtin_amdgcn_wmma_f32_16x16x4_f32(
      /*neg_a=*/false, a, /*neg_b=*/false, b,
      /*c_mod=*/(short)0, c, /*reuse_a=*/false, /*reuse_b=*/false);
  float r = d[0] + d[1] + d[2] + d[3] + d[4] + d[5] + d[6] + d[7];
  r += __shfl_xor(r, 16, 32);
  return r;
}

__global__ void __launch_bounds__(512)
recall_init_ws(float* ws_sum, unsigned int* ws_cnt) {
  if (threadIdx.x == 0) { *ws_sum = 0.0f; *ws_cnt = 0u; }
}

__global__ void __launch_bounds__(512)
recall_reduce(const float* __restrict__ pred, const int* __restrict__ gt,
              float* __restrict__ ws_sum, unsigned int* __restrict__ ws_cnt) {
  const v4f* __restrict__ predv = (const v4f*)pred;
  const v4i* __restrict__ gtv   = (const v4i*)gt;

  float s   = 0.0f;
  int   cnt = 0;

  const unsigned stride = gridDim.x * blockDim.x;   // NV_ is a multiple of this
  for (unsigned v = blockIdx.x * blockDim.x + threadIdx.x; v < NV_; v += stride) {
    const unsigned t4  = v & (TV_ - 1);             // vector column within row
    const unsigned row = v >> 9;                    // v / TV_
    const unsigned f   = row & (F_ - 1);
    const unsigned bb  = row >> 9;                  // row / F_
    const unsigned fp  = (f == 0) ? 1u : f;         // row-0 -> row-1 remap

    v4i g = __builtin_nontemporal_load(&gtv[v]);
    v4f p = __builtin_nontemporal_load(&predv[(bb * F_ + fp) * TV_ + t4]);
    if (t4 == 0) p[0] = p[1];                       // col-0 -> col-1 remap

    if (g[0]) { s += p[0]; ++cnt; }
    if (g[1]) { s += p[1]; ++cnt; }
    if (g[2]) { s += p[2]; ++cnt; }
    if (g[3]) { s += p[3]; ++cnt; }
  }

  // Intra-wave reduction via WMMA (all waves fully active: blockDim=512,
  // loop trip count is uniform across the grid).
  float wsum = wave_sum_wmma(s);
  float wcnt = wave_sum_wmma((float)cnt);           // <= 32*32*4 per wave: exact

  __shared__ float lsum[16];
  __shared__ float lcnt[16];
  const unsigned lane = threadIdx.x & 31u;
  const unsigned wave = threadIdx.x >> 5;           // 16 waves per block
  if (lane == 0) { lsum[wave] = wsum; lcnt[wave] = wcnt; }
  __syncthreads();

  if (wave == 0) {                                  // uniform per wave: EXEC all 1s
    float bs = (lane < 16u) ? lsum[lane] : 0.0f;    // select, not divergence
    float bc = (lane < 16u) ? lcnt[lane] : 0.0f;
    bs = wave_sum_wmma(bs);
    bc = wave_sum_wmma(bc);                         // <= 16384: exact in f32
    if (lane == 0) {
      atomicAdd(ws_sum, bs);
      atomicAdd(ws_cnt, (unsigned int)(bc + 0.5f));
    }
  }
}

__global__ void __launch_bounds__(512)
recall_finalize(const float* __restrict__ ws_sum,
                const unsigned int* __restrict__ ws_cnt,
                float* __restrict__ out) {
  if (threadIdx.x == 0) {
    const unsigned int c = *ws_cnt;
    out[0] = (c != 0u) ? (*ws_sum / (float)c) : 0.0f;
  }
}

extern "C" void kernel_launch(void* const* d_in, const int* in_sizes, int n_in,
                              void* d_out, int out_size, void* d_ws, size_t ws_size,
                              hipStream_t stream) {
  (void)in_sizes; (void)n_in; (void)out_size; (void)ws_size;
  const float* pred = (const float*)d_in[0];   // predicted: fp32 [32,512,2048]
  const int*   gt   = (const int*)d_in[1];     // gt: int32  [32,512,2048]
  float*        ws_sum = (float*)d_ws;                         // ws[0..3]
  unsigned int* ws_cnt = (unsigned int*)((char*)d_ws + 4);     // ws[4..7]
  float* out = (float*)d_out;

  recall_init_ws<<<1, 512, 0, stream>>>(ws_sum, ws_cnt);

  // 2048 blocks x 512 threads -> 8 float4+int4 iterations per thread;
  // pure b128 NT streaming of 268 MB (~11.5 us at 23.3 TB/s).
  recall_reduce<<<2048, 512, 0, stream>>>(pred, gt, ws_sum, ws_cnt);

  recall_finalize<<<1, 512, 0, stream>>>(ws_sum, ws_cnt, out);
}